// NetworkCBF_6871947674332
// MI455X (gfx1250) — compile-verified
//
#include <hip/hip_runtime.h>
#include <hip/hip_bf16.h>
#include <stdint.h>

#define NAG  4096
#define TOPK 12
#define NP   (NAG * TOPK)   // 49152 points

typedef __attribute__((ext_vector_type(2))) float v2f;
typedef __attribute__((ext_vector_type(8))) float v8f;

// ---------------------------------------------------------------------------
// Phase 1: per-row exact top-12 selection + feature build.
// One wave32 per row. Key = (bits(d2) << 12) | j  (d2 > 0 => monotone bits;
// j in [0,4096) fits 12 bits; lexicographic min == jax top_k tie-breaking).
// ---------------------------------------------------------------------------
__global__ __launch_bounds__(256) void topk_feat_kernel(
    const float* __restrict__ x, const float* __restrict__ r,
    float* __restrict__ out, float* __restrict__ feat)
{
    const int lane = threadIdx.x & 31;
    const int wv   = threadIdx.x >> 5;
    const int i    = blockIdx.x * 8 + wv;
    const size_t rowbase = (size_t)i * NAG;
    const float2* x2 = reinterpret_cast<const float2*>(x);

    unsigned long long best[TOPK];
#pragma unroll
    for (int v = 0; v < TOPK; ++v) best[v] = ~0ull;

    // scan 128 candidates per lane, keep sorted top-12
#pragma unroll 4
    for (int s = 0; s < NAG / 32; ++s) {
        int j = s * 32 + lane;
        float2 p = x2[(rowbase + (size_t)j) * 2];
        float d2 = sqrtf((p.x * p.x + 1e-6f) + (p.y * p.y + 1e-6f));
        unsigned long long key =
            ((unsigned long long)__float_as_uint(d2) << 12) | (unsigned)j;
        if (key < best[TOPK - 1]) {
            best[TOPK - 1] = key;
#pragma unroll
            for (int v = TOPK - 1; v >= 1; --v) {
                unsigned long long a = best[v - 1], b = best[v];
                unsigned long long lo = (a < b) ? a : b;
                unsigned long long hi = (a < b) ? b : a;
                best[v - 1] = lo;
                best[v]     = hi;
            }
        }
    }

    // merge 32 sorted lists: 12 rounds of wave-wide u64 min
    unsigned long long mykey = ~0ull;
#pragma unroll
    for (int k = 0; k < TOPK; ++k) {
        unsigned long long m = best[0];
#pragma unroll
        for (int off = 16; off >= 1; off >>= 1) {
            unsigned long long o = __shfl_xor(m, off, 32);
            m = (o < m) ? o : m;
        }
        if (best[0] == m) {           // unique winner (j unique per row)
#pragma unroll
            for (int v = 0; v < TOPK - 1; ++v) best[v] = best[v + 1];
            best[TOPK - 1] = ~0ull;
        }
        if (lane == k) mykey = m;
    }

    // lanes 0..11: gather features for neighbor k = lane
    if (lane < TOPK) {
        int j = (int)(mykey & 0xFFFull);
        const float4* x4 = reinterpret_cast<const float4*>(x);
        float4 xv = x4[rowbase + (size_t)j];
        int p = i * TOPK + lane;
        feat[0 * NP + p] = xv.x;
        feat[1 * NP + p] = xv.y;
        feat[2 * NP + p] = xv.z;
        feat[3 * NP + p] = xv.w;
        feat[4 * NP + p] = (j == i) ? 1.0f : 0.0f;
        float dn = sqrtf((xv.x * xv.x + 1e-4f) + (xv.y * xv.y + 1e-4f));
        feat[5 * NP + p] = dn - r[0];
        float mk = (dn <= 1.0f) ? 1.0f : 0.0f;
        out[NP + p]     = mk;          // mask region
        out[2 * NP + p] = (float)j;    // idx region (as float)
    }
}

// ---------------------------------------------------------------------------
// Phase 2: MLP 6->64->128->64->1 with V_WMMA_F32_16X16X4_F32.
// One wave per 16 points. LDS stride 20 dwords: 8*20 % 64 == 32, so the
// lane<16 / lane>=16 address sets hit disjoint bank groups.
// ---------------------------------------------------------------------------
#define HSTR 20

__device__ __forceinline__ v2f wA(const float* __restrict__ W, int Kdim,
                                  int m, int k0, int hi) {
    v2f a;
    a.x = W[m * Kdim + k0 + 2 * hi];
    a.y = W[m * Kdim + k0 + 2 * hi + 1];
    return a;
}
__device__ __forceinline__ v2f wB(const float* __restrict__ H, int k0,
                                  int hi, int p16) {
    v2f b;
    b.x = H[(k0 + 2 * hi) * HSTR + p16];
    b.y = H[(k0 + 2 * hi + 1) * HSTR + p16];
    return b;
}

__global__ __launch_bounds__(128) void mlp_wmma_kernel(
    const float* __restrict__ feat,
    const float* __restrict__ w1, const float* __restrict__ b1,
    const float* __restrict__ w2, const float* __restrict__ b2,
    const float* __restrict__ w3, const float* __restrict__ b3,
    const float* __restrict__ w4, const float* __restrict__ b4,
    float* __restrict__ out)
{
    __shared__ float h1s[4][64 * HSTR];    // also reused for H3
    __shared__ float h2s[4][128 * HSTR];

    const int lane = threadIdx.x & 31;
    const int wv   = threadIdx.x >> 5;
    const int hi   = lane >> 4;        // 0: lanes 0-15, 1: lanes 16-31
    const int p16  = lane & 15;        // point within 16-chunk
    const int p0   = (blockIdx.x * 4 + wv) * 16;
    float* H1 = h1s[wv];
    float* H2 = h2s[wv];

    // ---- layer 1: 6 -> 64 (K padded to 8) ----
    v2f bl0, bl1;
    {
        int k = 2 * hi;                                  // 0 or 2
        bl0.x = feat[(k)     * NP + p0 + p16];
        bl0.y = feat[(k + 1) * NP + p0 + p16];
        int k2 = 4 + 2 * hi;                             // 4 or 6
        bl1.x = (k2     < 6) ? feat[(k2)     * NP + p0 + p16] : 0.0f;
        bl1.y = (k2 + 1 < 6) ? feat[(k2 + 1) * NP + p0 + p16] : 0.0f;
    }
#pragma unroll
    for (int t = 0; t < 4; ++t) {
        int m = t * 16 + p16;
        v2f a0, a1;
        { int k = 2 * hi;     a0.x = w1[m * 6 + k];  a0.y = w1[m * 6 + k + 1]; }
        { int k = 4 + 2 * hi; a1.x = (k     < 6) ? w1[m * 6 + k]     : 0.0f;
                              a1.y = (k + 1 < 6) ? w1[m * 6 + k + 1] : 0.0f; }
        v8f c;
#pragma unroll
        for (int v = 0; v < 8; ++v) c[v] = b1[t * 16 + v + 8 * hi];
        c = __builtin_amdgcn_wmma_f32_16x16x4_f32(false, a0, false, bl0,
                                                  (short)0, c, false, false);
        c = __builtin_amdgcn_wmma_f32_16x16x4_f32(false, a1, false, bl1,
                                                  (short)0, c, false, false);
#pragma unroll
        for (int v = 0; v < 8; ++v)
            H1[(t * 16 + v + 8 * hi) * HSTR + p16] = fmaxf(c[v], 0.0f);
    }
    __syncthreads();

    // ---- layer 2: 64 -> 128 ----
#pragma unroll
    for (int t = 0; t < 8; ++t) {
        int m = t * 16 + p16;
        v8f c;
#pragma unroll
        for (int v = 0; v < 8; ++v) c[v] = b2[t * 16 + v + 8 * hi];
#pragma unroll
        for (int s = 0; s < 16; ++s) {
            v2f a = wA(w2, 64, m, 4 * s, hi);
            v2f b = wB(H1, 4 * s, hi, p16);
            c = __builtin_amdgcn_wmma_f32_16x16x4_f32(false, a, false, b,
                                                      (short)0, c, false, false);
        }
#pragma unroll
        for (int v = 0; v < 8; ++v)
            H2[(t * 16 + v + 8 * hi) * HSTR + p16] = fmaxf(c[v], 0.0f);
    }
    __syncthreads();

    // ---- layer 3: 128 -> 64 (write into H1 storage) ----
#pragma unroll
    for (int t = 0; t < 4; ++t) {
        int m = t * 16 + p16;
        v8f c;
#pragma unroll
        for (int v = 0; v < 8; ++v) c[v] = b3[t * 16 + v + 8 * hi];
#pragma unroll
        for (int s = 0; s < 32; ++s) {
            v2f a = wA(w3, 128, m, 4 * s, hi);
            v2f b = wB(H2, 4 * s, hi, p16);
            c = __builtin_amdgcn_wmma_f32_16x16x4_f32(false, a, false, b,
                                                      (short)0, c, false, false);
        }
#pragma unroll
        for (int v = 0; v < 8; ++v)
            H1[(t * 16 + v + 8 * hi) * HSTR + p16] = fmaxf(c[v], 0.0f);
    }
    __syncthreads();

    // ---- layer 4: 64 -> 1, apply mask ----
    if (lane < 16) {
        int p = p0 + lane;
        float acc = b4[0];
#pragma unroll 8
        for (int c = 0; c < 64; ++c)
            acc = fmaf(w4[c], H1[c * HSTR + lane], acc);
        float mk = out[NP + p];
        out[p] = acc * mk;
    }
}

// ---------------------------------------------------------------------------
extern "C" void kernel_launch(void* const* d_in, const int* in_sizes, int n_in,
                              void* d_out, int out_size, void* d_ws, size_t ws_size,
                              hipStream_t stream) {
    const float* x  = (const float*)d_in[0];
    const float* r  = (const float*)d_in[1];
    const float* w1 = (const float*)d_in[2];
    const float* b1 = (const float*)d_in[3];
    const float* w2 = (const float*)d_in[4];
    const float* b2 = (const float*)d_in[5];
    const float* w3 = (const float*)d_in[6];
    const float* b3 = (const float*)d_in[7];
    const float* w4 = (const float*)d_in[8];
    const float* b4 = (const float*)d_in[9];
    float* out  = (float*)d_out;
    float* feat = (float*)d_ws;   // 6 * 49152 floats = 1.125 MB scratch

    topk_feat_kernel<<<NAG / 8, 256, 0, stream>>>(x, r, out, feat);
    mlp_wmma_kernel<<<NP / 64, 128, 0, stream>>>(feat, w1, b1, w2, b2,
                                                 w3, b3, w4, b4, out);
}